// WordNetAllEmbedding_27960237097278
// MI455X (gfx1250) — compile-verified
//
#include <hip/hip_runtime.h>
#include <hip/hip_bf16.h>

// ---------------------------------------------------------------------------
// out[r, :200] = entity_table[ids[r]] (512) @ W[0:512, :] + P[pos_index[ids[r]]]
// where P[j] = pos_table[j] @ W[512:537, :] + b   (only 9 distinct rows)
// Computed with V_WMMA_F32_16X16X4_F32 (full f32 precision, f32 accumulate).
// ---------------------------------------------------------------------------

typedef __attribute__((ext_vector_type(2))) float v2f;
typedef __attribute__((ext_vector_type(8))) float v8f;

#define DE     512
#define DOUT   200
#define NT     13          // ceil(200/16) N-tiles (padded to 208)
#define NPAD   208
#define NPOS   9
#define KSTEPS 128         // 512 / 4
#define M_TOT  (32*128*30) // 122880 rows

// Pack W[0:512, :200] into WMMA-B-fragment order:
// Wp[((t*128 + s)*32 + lane)] = { W[4s+2*(lane>>4)    ][16t + (lane&15)],
//                                 W[4s+2*(lane>>4) + 1][16t + (lane&15)] }
// (zero for padded columns >= 200). One coalesced b64 per lane in the GEMM.
__global__ __launch_bounds__(256) void pack_w_kernel(const float* __restrict__ W,
                                                     float2* __restrict__ Wp) {
  int idx = blockIdx.x * 256 + threadIdx.x;
  if (idx >= NT * KSTEPS * 32) return;
  int l = idx & 31;
  int s = (idx >> 5) & (KSTEPS - 1);
  int t = idx >> 12;
  int col = t * 16 + (l & 15);
  int k0  = 4 * s + ((l >> 4) << 1);
  float2 v; v.x = 0.f; v.y = 0.f;
  if (col < DOUT) {
    v.x = W[(size_t)k0 * DOUT + col];
    v.y = W[(size_t)(k0 + 1) * DOUT + col];
  }
  Wp[idx] = v;
}

__global__ __launch_bounds__(256) void wordnet_proj_kernel(
    const int*   __restrict__ ids,     // (M_TOT,)
    const float* __restrict__ etab,    // (V, 512)
    const float* __restrict__ ptab,    // (V, 25), rows 0..8 used
    const int*   __restrict__ pindex,  // (V,)
    const float* __restrict__ W,       // (537, 200)
    const float* __restrict__ bias,    // (200,)
    const float2* __restrict__ Wp,     // packed W (or unused)
    int use_packed,
    float* __restrict__ out)           // (M_TOT, 200)
{
  // --- per-block: 9 x 208 POS projections (incl. bias) into LDS -----------
  __shared__ float Pls[NPOS * NPAD];
  for (int e = threadIdx.x; e < NPOS * NPAD; e += 256) {
    int j = e / NPAD, col = e % NPAD;
    float acc = 0.f;
    if (col < DOUT) {
      acc = bias[col];
      #pragma unroll
      for (int kk = 0; kk < 25; ++kk)
        acc += ptab[j * 25 + kk] * W[(size_t)(DE + kk) * DOUT + col];
    }
    Pls[e] = acc;
  }
  __syncthreads();

  const int lane = threadIdx.x & 31;
  const int wave = threadIdx.x >> 5;
  const int hi   = lane >> 4;      // 0: K=0..1 half, 1: K=2..3 half
  const int ln   = lane & 15;
  const int row_base = (blockIdx.x * 8 + wave) * 16;   // 16-row M tile per wave

  // A gather pointer: lane holds row (row_base+ln), K offset 2*hi within each K=4 step
  const int aid = ids[row_base + ln];
  const float* aptr = etab + (size_t)aid * DE + (hi << 1);

  // POS row index for each of the 8 output rows this lane contributes to
  int pid8[8];
  #pragma unroll
  for (int i = 0; i < 8; ++i)
    pid8[i] = pindex[ids[row_base + hi * 8 + i]];

  v8f acc[NT];
  #pragma unroll
  for (int t = 0; t < NT; ++t) acc[t] = (v8f){0.f,0.f,0.f,0.f,0.f,0.f,0.f,0.f};

  if (use_packed) {
    const v2f* __restrict__ wp = (const v2f*)Wp;
    for (int s = 0; s < KSTEPS; ++s) {
      v2f a = *(const v2f*)(aptr + 4 * s);
      __builtin_prefetch(aptr + 4 * (s + 8), 0, 0);   // global_prefetch_b8, A stream
      #pragma unroll
      for (int t = 0; t < NT; ++t) {
        v2f bf = wp[(size_t)(t * KSTEPS + s) * 32 + lane];
        acc[t] = __builtin_amdgcn_wmma_f32_16x16x4_f32(
            false, a, false, bf, (short)0, acc[t], false, false);
      }
    }
  } else {
    for (int s = 0; s < KSTEPS; ++s) {
      v2f a = *(const v2f*)(aptr + 4 * s);
      int k0 = 4 * s + (hi << 1);
      #pragma unroll
      for (int t = 0; t < NT; ++t) {
        int col = t * 16 + ln;
        v2f bf = (v2f){0.f, 0.f};
        if (col < DOUT) {
          bf.x = W[(size_t)k0 * DOUT + col];
          bf.y = W[(size_t)(k0 + 1) * DOUT + col];
        }
        acc[t] = __builtin_amdgcn_wmma_f32_16x16x4_f32(
            false, a, false, bf, (short)0, acc[t], false, false);
      }
    }
  }

  // --- epilogue: D[M][N] layout: vgpr i -> row = row_base + hi*8 + i, col = ln
  #pragma unroll
  for (int t = 0; t < NT; ++t) {
    int col = t * 16 + ln;
    if (col < DOUT) {
      #pragma unroll
      for (int i = 0; i < 8; ++i) {
        int r = row_base + hi * 8 + i;
        out[(size_t)r * DOUT + col] = acc[t][i] + Pls[pid8[i] * NPAD + col];
      }
    }
  }
}

extern "C" void kernel_launch(void* const* d_in, const int* in_sizes, int n_in,
                              void* d_out, int out_size, void* d_ws, size_t ws_size,
                              hipStream_t stream) {
  const int*   entity_ids   = (const int*)  d_in[0];   // (32,128,30) int32
  const float* entity_table = (const float*)d_in[1];   // (400000,512) f32
  const float* pos_table    = (const float*)d_in[2];   // (400000,25)  f32
  const int*   pos_index    = (const int*)  d_in[3];   // (400000,)    int32
  const float* W            = (const float*)d_in[4];   // (537,200)    f32
  const float* b            = (const float*)d_in[5];   // (200,)       f32
  float* out = (float*)d_out;                          // (32,128,30,200) f32

  const size_t wp_bytes = (size_t)NT * KSTEPS * 32 * 2 * sizeof(float); // ~426 KB
  const int use_packed = (d_ws != nullptr) && (ws_size >= wp_bytes);
  float2* Wp = (float2*)d_ws;

  if (use_packed) {
    int n = NT * KSTEPS * 32;
    pack_w_kernel<<<(n + 255) / 256, 256, 0, stream>>>(W, Wp);
  }

  // 960 blocks * 8 waves * 16 rows = 122880 = M_TOT
  wordnet_proj_kernel<<<M_TOT / 128, 256, 0, stream>>>(
      entity_ids, entity_table, pos_table, pos_index, W, b, Wp, use_packed, out);
}